// GCN_plus_GAP_Model_79680233276324
// MI455X (gfx1250) — compile-verified
//
#include <hip/hip_runtime.h>
#include <hip/hip_bf16.h>

typedef __attribute__((ext_vector_type(2))) float v2f;
typedef __attribute__((ext_vector_type(8))) float v8f;

// ---------------------------------------------------------------------------
// fp32 atomic add -> native global_atomic_add_f32
// ---------------------------------------------------------------------------
__device__ __forceinline__ void atomAddF(float* p, float v) {
    unsafeAtomicAdd(p, v);
}

// ---------------------------------------------------------------------------
// fill
// ---------------------------------------------------------------------------
__global__ void fill_f32(float* __restrict__ p, float v, int n) {
    int i = blockIdx.x * blockDim.x + threadIdx.x;
    if (i < n) p[i] = v;
}

// ---------------------------------------------------------------------------
// degree: deg[dst[e]] += 1
// ---------------------------------------------------------------------------
__global__ void deg_kernel(const int* __restrict__ dst, float* __restrict__ deg, int E) {
    int e = blockIdx.x * blockDim.x + threadIdx.x;
    if (e < E) atomAddF(&deg[dst[e]], 1.0f);
}

// dinv = rsqrt(deg + 1)   (in place)
__global__ void dinv_kernel(float* __restrict__ d, int N) {
    int i = blockIdx.x * blockDim.x + threadIdx.x;
    if (i < N) d[i] = __frsqrt_rn(d[i] + 1.0f);
}

// norm[e] = dinv[src]*dinv[dst]   (layer-invariant)
__global__ void norm_kernel(const int* __restrict__ src, const int* __restrict__ dst,
                            const float* __restrict__ dinv, float* __restrict__ norm, int E) {
    int e = blockIdx.x * blockDim.x + threadIdx.x;
    if (e < E) norm[e] = dinv[src[e]] * dinv[dst[e]];
}

// ---------------------------------------------------------------------------
// xw = h @ W^T  via V_WMMA_F32_16X16X4_F32.  One wave -> 16x64 output tile.
// A frag (16x4 f32):  lane<16: (M=lane, K=k0,k0+1); lane>=16: (M=lane-16, K=k0+2,k0+3)
// B frag (4x16 f32):  B[k][j] = W[j][k]; lane supplies column j = n*16 + (lane&15)
// D (16x16 f32): vgpr r: lanes0-15 -> M=r, lanes16-31 -> M=r+8, N = lane&15
// ---------------------------------------------------------------------------
__global__ __launch_bounds__(256) void gemm64_wmma(const float* __restrict__ hin,
                                                   const float* __restrict__ W,
                                                   float* __restrict__ out, int N) {
    __shared__ float lw[64 * 66];                 // stride-66 pad kills bank conflicts
    int tid = threadIdx.x;
    for (int idx = tid; idx < 64 * 64; idx += 256) {
        int r = idx >> 6, k = idx & 63;
        lw[r * 66 + k] = W[idx];
    }
    __syncthreads();

    int lane  = tid & 31;
    int wave  = tid >> 5;
    int tile  = blockIdx.x * 8 + wave;
    int ntile = (N + 15) >> 4;
    if (tile >= ntile) return;                    // wave-uniform: EXEC stays all-ones

    int row0 = tile << 4;
    int m    = lane & 15;
    int hk   = (lane >> 4) << 1;                  // 0 or 2
    int rowa = row0 + m;
    if (rowa > N - 1) rowa = N - 1;               // clamp loads, guard stores later
    const float* hrow = hin + (size_t)rowa * 64;

    v8f acc[4] = {};
#pragma unroll
    for (int ks = 0; ks < 16; ++ks) {
        int k = (ks << 2) + hk;
        v2f a;
        a.x = hrow[k];
        a.y = hrow[k + 1];
#pragma unroll
        for (int n = 0; n < 4; ++n) {
            const float* wp = &lw[(n * 16 + m) * 66 + k];
            v2f b;
            b.x = wp[0];
            b.y = wp[1];
            acc[n] = __builtin_amdgcn_wmma_f32_16x16x4_f32(
                false, a, false, b, (short)0, acc[n], false, false);
        }
    }

    int mo = (lane >> 4) << 3;                    // 0 or 8
#pragma unroll
    for (int n = 0; n < 4; ++n) {
#pragma unroll
        for (int r = 0; r < 8; ++r) {
            int row = row0 + mo + r;
            if (row < N) out[(size_t)row * 64 + n * 16 + m] = acc[n][r];
        }
    }
}

// ---------------------------------------------------------------------------
// agg init: agg = xw * dinv^2 + b   (agg lives in d_out slice)
// ---------------------------------------------------------------------------
__global__ void init_agg(const float* __restrict__ xw, const float* __restrict__ dinv,
                         const float* __restrict__ b, float* __restrict__ agg, int NE) {
    int t = blockIdx.x * blockDim.x + threadIdx.x;
    if (t >= NE) return;
    int i = t >> 6, f = t & 63;
    float di = dinv[i];
    agg[t] = xw[t] * di * di + b[f];
}

// ---------------------------------------------------------------------------
// edge scatter: 16 threads/edge, float4 gather + 4 fp32 atomics
// ---------------------------------------------------------------------------
__global__ void edge_scatter(const int* __restrict__ src, const int* __restrict__ dst,
                             const float* __restrict__ norm, const float* __restrict__ xw,
                             float* __restrict__ agg, int E) {
    int t = blockIdx.x * blockDim.x + threadIdx.x;
    int e = t >> 4;
    if (e >= E) return;
    int f = (t & 15) << 2;
    int s = src[e], d = dst[e];
    float nw = norm[e];
    const float4 v = *(const float4*)(xw + (size_t)s * 64 + f);
    float* o = agg + (size_t)d * 64 + f;
    atomAddF(o + 0, v.x * nw);
    atomAddF(o + 1, v.y * nw);
    atomAddF(o + 2, v.z * nw);
    atomAddF(o + 3, v.w * nw);
}

// relu in place
__global__ void relu_inplace(float* __restrict__ p, int n) {
    int i = blockIdx.x * blockDim.x + threadIdx.x;
    if (i < n) p[i] = fmaxf(p[i], 0.0f);
}

// ---------------------------------------------------------------------------
// pooling
// ---------------------------------------------------------------------------
__global__ void count_kernel(const int* __restrict__ batch, float* __restrict__ counts, int N) {
    int i = blockIdx.x * blockDim.x + threadIdx.x;
    if (i < N) atomAddF(&counts[batch[i]], 1.0f);
}

__global__ void pool_scatter(const float* __restrict__ h3, const int* __restrict__ batch,
                             float* __restrict__ pooled, int N) {
    int t = blockIdx.x * blockDim.x + threadIdx.x;
    int i = t >> 4;
    if (i >= N) return;
    int f = (t & 15) << 2;
    int g = batch[i];
    const float4 v = *(const float4*)(h3 + (size_t)i * 64 + f);
    float* o = pooled + (size_t)g * 64 + f;
    atomAddF(o + 0, v.x);
    atomAddF(o + 1, v.y);
    atomAddF(o + 2, v.z);
    atomAddF(o + 3, v.w);
}

// ---------------------------------------------------------------------------
// finish: divide pooled (in place), FFN + relu, softmax.  One thread / graph.
// ---------------------------------------------------------------------------
__global__ void ffn_finish(float* __restrict__ pooled, const float* __restrict__ counts,
                           const float* __restrict__ Wf, const float* __restrict__ bf,
                           float* __restrict__ ffn, float* __restrict__ soft, int G, int C) {
    int g = blockIdx.x * blockDim.x + threadIdx.x;
    if (g >= G) return;
    float inv = 1.0f / fmaxf(counts[g], 1.0f);
    float p[64];
#pragma unroll
    for (int h = 0; h < 64; ++h) {
        float v = pooled[(size_t)g * 64 + h] * inv;
        p[h] = v;
        pooled[(size_t)g * 64 + h] = v;
    }
    float l[16];
    float mx = -1e30f;
    for (int c = 0; c < C; ++c) {
        float acc = bf[c];
        const float* wr = Wf + (size_t)c * 64;
#pragma unroll
        for (int h = 0; h < 64; ++h) acc += p[h] * wr[h];
        acc = fmaxf(acc, 0.0f);
        l[c] = acc;
        ffn[(size_t)g * C + c] = acc;
        mx = fmaxf(mx, acc);
    }
    float s = 0.0f;
    for (int c = 0; c < C; ++c) {
        float e = __expf(l[c] - mx);
        l[c] = e;
        s += e;
    }
    float is = 1.0f / s;
    for (int c = 0; c < C; ++c) soft[(size_t)g * C + c] = l[c] * is;
}

// ---------------------------------------------------------------------------
// launch
// ---------------------------------------------------------------------------
static inline int cdiv(long long a, int b) { return (int)((a + b - 1) / b); }

extern "C" void kernel_launch(void* const* d_in, const int* in_sizes, int n_in,
                              void* d_out, int out_size, void* d_ws, size_t ws_size,
                              hipStream_t stream) {
    const float* x     = (const float*)d_in[0];
    const int*   ei    = (const int*)d_in[1];
    const int*   batch = (const int*)d_in[2];
    const float* W1 = (const float*)d_in[3];
    const float* b1 = (const float*)d_in[4];
    const float* W2 = (const float*)d_in[5];
    const float* b2 = (const float*)d_in[6];
    const float* W3 = (const float*)d_in[7];
    const float* b3 = (const float*)d_in[8];
    const float* Wf = (const float*)d_in[9];
    const float* bf = (const float*)d_in[10];

    const int N = in_sizes[0] / 64;
    const int E = in_sizes[1] / 2;
    const int C = in_sizes[9] / 64;
    const int G = (out_size - 3 * N * 64) / (64 + 2 * C);

    const int* src = ei;
    const int* dst = ei + E;

    // d_out slices
    float* h1      = (float*)d_out;
    float* h2      = h1 + (size_t)N * 64;
    float* h3      = h2 + (size_t)N * 64;
    float* pooled  = h3 + (size_t)N * 64;
    float* ffn     = pooled + (size_t)G * 64;
    float* soft    = ffn + (size_t)G * C;

    // workspace slices
    float* dinv   = (float*)d_ws;
    float* norm   = dinv + N;
    float* xw     = norm + E;
    float* counts = xw + (size_t)N * 64;

    const int NE  = N * 64;
    const int TB  = 256;
    const int gTiles  = cdiv((N + 15) / 16, 8);

    // degrees + per-edge norm (once per call)
    fill_f32<<<cdiv(N, TB), TB, 0, stream>>>(dinv, 0.0f, N);
    deg_kernel<<<cdiv(E, TB), TB, 0, stream>>>(dst, dinv, E);
    dinv_kernel<<<cdiv(N, TB), TB, 0, stream>>>(dinv, N);
    norm_kernel<<<cdiv(E, TB), TB, 0, stream>>>(src, dst, dinv, norm, E);

    const float* hin = x;
    float*       hl[3] = {h1, h2, h3};
    const float* Ws[3] = {W1, W2, W3};
    const float* bs[3] = {b1, b2, b3};
    for (int layer = 0; layer < 3; ++layer) {
        gemm64_wmma<<<gTiles, TB, 0, stream>>>(hin, Ws[layer], xw, N);
        init_agg<<<cdiv(NE, TB), TB, 0, stream>>>(xw, dinv, bs[layer], hl[layer], NE);
        edge_scatter<<<cdiv((long long)E * 16, TB), TB, 0, stream>>>(src, dst, norm, xw,
                                                                     hl[layer], E);
        relu_inplace<<<cdiv(NE, TB), TB, 0, stream>>>(hl[layer], NE);
        hin = hl[layer];
    }

    // global mean pool
    fill_f32<<<cdiv(G * 64, TB), TB, 0, stream>>>(pooled, 0.0f, G * 64);
    fill_f32<<<cdiv(G, TB), TB, 0, stream>>>(counts, 0.0f, G);
    count_kernel<<<cdiv(N, TB), TB, 0, stream>>>(batch, counts, N);
    pool_scatter<<<cdiv((long long)N * 16, TB), TB, 0, stream>>>(h3, batch, pooled, N);

    // FFN + softmax
    ffn_finish<<<cdiv(G, TB), TB, 0, stream>>>(pooled, counts, Wf, bf, ffn, soft, G, C);
}